// AggFeatureModel_85761906967230
// MI455X (gfx1250) — compile-verified
//
#include <hip/hip_runtime.h>
#include <math.h>

// ---------------------------------------------------------------------------
// AggFeatureModel on MI455X (gfx1250).
// Memory-bound: ~31MB traffic -> ~1.3us roofline @ 23.3 TB/s. No matrix math;
// the CDNA5 win is TDM (tensor_load_to_lds) double-buffered global->LDS
// staging overlapped with the ds_add_u32 LDS histogram phase.
//
// Simplification vs reference: the categorical "values" are the codes, so
// msum = cnt*c, msq = cnt*c^2 -> only the count histogram is accumulated.
// ---------------------------------------------------------------------------

#define TT    1024
#define NMCC  200
#define NTR   100
#define NCOLS 906
#define EPSF  1e-9f

typedef __attribute__((ext_vector_type(4))) unsigned int v4u;
typedef __attribute__((ext_vector_type(8))) int          v8i;
typedef __attribute__((ext_vector_type(4))) int          v4i;

// Issue one TDM load of `ndw` dwords (1D tile) from global -> LDS.
// D# built per CDNA5 ISA ch.8: group0 = {count=1, lds_addr, global_addr,
// type=2}; group1 = {data_size=4B, tensor_dim0=ndw, tensor_dim1=1,
// tile_dim0=ndw, tile_dim1=1, dim0_stride=ndw}. Groups 2/3 zero (<=2D).
// Toolchain here is the 6-arg clang-23 form:
//   (uint32x4 g0, int32x8 g1, int32x4, int32x4, int32x8, i32 cpol)
static __device__ __forceinline__ void tdm_load_dwords(unsigned lds_off,
                                                       const void* gp,
                                                       unsigned ndw) {
  unsigned long long ga = (unsigned long long)(uintptr_t)gp;
  v4u g0;
  g0.x = 1u;                                             // count=1 (valid)
  g0.y = lds_off;                                        // lds_addr (bytes)
  g0.z = (unsigned)(ga & 0xffffffffull);                 // global_addr[31:0]
  g0.w = (unsigned)((ga >> 32) & 0x1ffffffull)           // global_addr[56:32]
         | (2u << 30);                                   // type=2 ("image")
  v8i g1;
  g1[0] = (int)(2u << 16);          // data_size=2 -> 4 bytes
  g1[1] = (int)(ndw << 16);         // tensor_dim0[15:0]  @ bits[63:48]
  g1[2] = (int)(1u << 16);          // tensor_dim0 hi=0, tensor_dim1=1
  g1[3] = (int)(ndw << 16);         // tile_dim0          @ bits[127:112]
  g1[4] = 1;                        // tile_dim1=1, tile_dim2=0
  g1[5] = (int)ndw;                 // tensor_dim0_stride[31:0]
  g1[6] = 0;                        // stride hi, dim1_stride lo
  g1[7] = 0;
  v4i z4 = (v4i){0, 0, 0, 0};
  v8i z8 = (v8i){0, 0, 0, 0, 0, 0, 0, 0};
  __builtin_amdgcn_tensor_load_to_lds(g0, g1, z4, z4, z8, 0);
}

static __device__ __forceinline__ float wred(float v) {
#pragma unroll
  for (int o = 16; o > 0; o >>= 1) v += __shfl_down(v, o, 32);
  return v;
}

// Deterministic block-wide reduction of 4 floats (fixed tree order).
static __device__ __forceinline__ void breduce4(const float vin[4],
                                                float out[4],
                                                float (*s_red)[4], int tid) {
  float r0 = wred(vin[0]), r1 = wred(vin[1]);
  float r2 = wred(vin[2]), r3 = wred(vin[3]);
  int lane = tid & 31, w = tid >> 5;
  if (lane == 0) {
    s_red[w][0] = r0; s_red[w][1] = r1; s_red[w][2] = r2; s_red[w][3] = r3;
  }
  __syncthreads();
  if (tid == 0) {
    float a0 = 0.f, a1 = 0.f, a2 = 0.f, a3 = 0.f;
#pragma unroll
    for (int i = 0; i < 8; ++i) {
      a0 += s_red[i][0]; a1 += s_red[i][1];
      a2 += s_red[i][2]; a3 += s_red[i][3];
    }
    s_red[0][0] = a0; s_red[0][1] = a1; s_red[0][2] = a2; s_red[0][3] = a3;
  }
  __syncthreads();
  out[0] = s_red[0][0]; out[1] = s_red[0][1];
  out[2] = s_red[0][2]; out[3] = s_red[0][3];
  __syncthreads();  // scratch may be reused immediately after return
}

__global__ __launch_bounds__(256) void agg_feature_kernel(
    const float* __restrict__ amount, const int* __restrict__ mcc,
    const int* __restrict__ tr, const int* __restrict__ seq_lens,
    float* __restrict__ out, int nrows) {
  __shared__ __align__(16) float    s_amt[2][TT];
  __shared__ __align__(16) unsigned s_mcc[2][TT];
  __shared__ __align__(16) unsigned s_tr[2][TT];
  __shared__ unsigned s_hist_mcc[NMCC];
  __shared__ unsigned s_hist_tr[NTR];
  __shared__ float    s_red[8][4];

  const int tid  = threadIdx.x;
  const int wv   = tid >> 5;          // 8 waves of 32
  const int row0 = blockIdx.x * 2;    // 2 rows per block, double-buffered
  if (row0 >= nrows) return;

  // Each wave DMAs its own 128-dword slice of each of the 3 arrays, so every
  // wave tracks its own TENSORcnt; a workgroup barrier completes the handoff.
  const unsigned sl = (unsigned)wv * 128u;

  auto issue_row = [&](int row, int buf) {
    const size_t gbase = (size_t)row * TT + sl;
    tdm_load_dwords((unsigned)(uintptr_t)&s_amt[buf][sl], amount + gbase, 128u);
    tdm_load_dwords((unsigned)(uintptr_t)&s_mcc[buf][sl], mcc + gbase, 128u);
    tdm_load_dwords((unsigned)(uintptr_t)&s_tr[buf][sl],  tr + gbase,  128u);
  };

  issue_row(row0, 0);  // prime the pipeline

  for (int r = 0; r < 2; ++r) {
    const int row = row0 + r;
    if (row >= nrows) break;  // uniform per block

    // Zero histograms (disjoint from in-flight TDM tile writes).
    if (tid < NMCC) s_hist_mcc[tid] = 0u;
    if (tid < NTR)  s_hist_tr[tid]  = 0u;

    // Prefetch next row into the other buffer, then drain this row's 3 loads.
    if (r == 0) {
      if (row0 + 1 < nrows) {
        issue_row(row0 + 1, 1);                 // per-wave cnt: 3 -> 6
        __builtin_amdgcn_s_wait_tensorcnt(3);   // first 3 (row r) complete
      } else {
        __builtin_amdgcn_s_wait_tensorcnt(0);
      }
    } else {
      __builtin_amdgcn_s_wait_tensorcnt(0);
    }
    __syncthreads();  // all waves' slices + hist zeroing visible

    // ---- Phase B: LDS histogram (ds_add_u32) + amount partial sums ----
    const uint4  mv = ((const uint4*)&s_mcc[r & 1][0])[tid];
    const uint4  tv = ((const uint4*)&s_tr[r & 1][0])[tid];
    const float4 av = ((const float4*)&s_amt[r & 1][0])[tid];
    atomicAdd(&s_hist_mcc[mv.x], 1u); atomicAdd(&s_hist_mcc[mv.y], 1u);
    atomicAdd(&s_hist_mcc[mv.z], 1u); atomicAdd(&s_hist_mcc[mv.w], 1u);
    atomicAdd(&s_hist_tr[tv.x], 1u);  atomicAdd(&s_hist_tr[tv.y], 1u);
    atomicAdd(&s_hist_tr[tv.z], 1u);  atomicAdd(&s_hist_tr[tv.w], 1u);

    float ps = (av.x + av.y) + (av.z + av.w);
    float pq = (av.x * av.x + av.y * av.y) + (av.z * av.z + av.w * av.w);

    float vin[4] = {ps, pq, 0.f, 0.f}, S[4];
    breduce4(vin, S, s_red, tid);  // its barriers also fence the atomics

    float* __restrict__ orow = out + (size_t)row * NCOLS;
    if (tid == 0) {
      const float slen = (float)seq_lens[row];
      const float s = S[0], sq = S[1];
      orow[0] = slen;
      orow[1] = s;
      orow[2] = s / slen;
      orow[3] = sqrtf((sq - s * s / slen) / (slen - 1.0f + EPSF));
    }

    // ---- Phase C: per-bin features (one bin per thread; sizes <= 256) ----
#pragma unroll
    for (int cat = 0; cat < 2; ++cat) {
      const int       size = (cat == 0) ? NMCC : NTR;
      const unsigned* h    = (cat == 0) ? s_hist_mcc : s_hist_tr;
      const int       ob   = (cat == 0) ? 4 : 604;   // cnt block base col
      const int       dcol = (cat == 0) ? 904 : 905; // distinct-count col

      float e_cnt = 0.f, msum = 0.f, stdv = 0.f, flag = 0.f;
      if (tid < size) {
        const float n  = (float)h[tid];
        const float cf = (float)tid;
        e_cnt = (tid == 0) ? 0.0f : n;        // mask zeroes bin 0
        msum  = n * cf;                       // sum of code values
        const float msq = n * (cf * cf);      // sum of squared codes
        const float num = msq - msum * msum / (e_cnt + 1e-9f);
        const float den = fmaxf(e_cnt - 1.0f, 0.0f) + 1e-9f;
        stdv = sqrtf(num / den);
        flag = (e_cnt > 0.0f) ? 1.0f : 0.0f;
      }
      float vin2[4] = {e_cnt, msum, stdv, flag}, S2[4];
      breduce4(vin2, S2, s_red, tid);
      if (tid < size) {
        orow[ob + tid]            = e_cnt / (S2[0] + EPSF);
        orow[ob + size + tid]     = msum / (S2[1] + EPSF);
        orow[ob + 2 * size + tid] = stdv / (S2[2] + EPSF);
      }
      if (tid == 0) orow[dcol] = S2[3];
    }
    __syncthreads();  // hist reads done before next iter's zeroing/reuse
  }
}

extern "C" void kernel_launch(void* const* d_in, const int* in_sizes, int n_in,
                              void* d_out, int out_size, void* d_ws,
                              size_t ws_size, hipStream_t stream) {
  const float* amount = (const float*)d_in[0];
  const int*   mcc    = (const int*)d_in[1];
  const int*   tr     = (const int*)d_in[2];
  const int*   seq    = (const int*)d_in[3];
  float*       out    = (float*)d_out;
  const int B = in_sizes[3];          // seq_lens length
  const int grid = (B + 1) / 2;       // 2 rows per block (double-buffered)
  agg_feature_kernel<<<grid, 256, 0, stream>>>(amount, mcc, tr, seq, out, B);
}